// CondensationLoss_11209864642828
// MI455X (gfx1250) — compile-verified
//
#include <hip/hip_runtime.h>
#include <hip/hip_bf16.h>
#include <math.h>

#define N_HITS   40000
#define K_OBJ    1200
#define K_PAD    1216            // multiple of 16
#define ROW_TILES (N_HITS / 16)  // 2500
#define COL_TILES (K_PAD / 16)   // 76
#define D_DIM    16
#define XK_STRIDE 18             // padded LDS stride: 18*k mod 64 injective for k=0..15
#define NB_PAIR  256             // blocks for pair kernel
#define EPSF     1e-9f

typedef float v2f __attribute__((ext_vector_type(2)));
typedef float v8f __attribute__((ext_vector_type(8)));

// ---------------- workspace layout ----------------
struct WS {
    unsigned long long* packed;  // [K_PAD] (q_bits<<32)|(~i)
    unsigned int*       cnt;     // [K_PAD]
    unsigned int*       alpha;   // [K_PAD]
    float*              q;       // [N]
    float*              xn;      // [N]  ||x_i||^2
    float*              xk;      // [K_PAD*16]
    float*              qk;      // [K_PAD]
    float*              xkn;     // [K_PAD]
    float*              inv_att; // [K_PAD]
    float*              inv_rep; // [K_PAD]
    float*              att_part;// [NB_PAIR]
    float*              rep_part;// [NB_PAIR]
};

static inline size_t align256(size_t v) { return (v + 255) & ~size_t(255); }

static WS carve(void* base) {
    char* p = (char*)base;
    WS w;
    w.packed  = (unsigned long long*)p; p += align256(K_PAD * 8);
    w.cnt     = (unsigned int*)p;       p += align256(K_PAD * 4);
    w.alpha   = (unsigned int*)p;       p += align256(K_PAD * 4);
    w.q       = (float*)p;              p += align256(N_HITS * 4);
    w.xn      = (float*)p;              p += align256(N_HITS * 4);
    w.xk      = (float*)p;              p += align256(K_PAD * D_DIM * 4);
    w.qk      = (float*)p;              p += align256(K_PAD * 4);
    w.xkn     = (float*)p;              p += align256(K_PAD * 4);
    w.inv_att = (float*)p;              p += align256(K_PAD * 4);
    w.inv_rep = (float*)p;              p += align256(K_PAD * 4);
    w.att_part= (float*)p;              p += align256(NB_PAIR * 4);
    w.rep_part= (float*)p;              p += align256(NB_PAIR * 4);
    return w;
}

// ---------------- kernel 1: init scratch ----------------
__global__ void k_init(unsigned long long* packed, unsigned int* cnt) {
    int k = blockIdx.x * blockDim.x + threadIdx.x;
    if (k < K_PAD) { packed[k] = 0ull; cnt[k] = 0u; }
}

// ---------------- kernel 2: per-hit ----------------
__global__ void k_hit(const float* __restrict__ beta,
                      const float* __restrict__ x,
                      const int*   __restrict__ oid,
                      unsigned long long* __restrict__ packed,
                      unsigned int* __restrict__ cnt,
                      float* __restrict__ q, float* __restrict__ xn) {
    int i = blockIdx.x * blockDim.x + threadIdx.x;
    if (i >= N_HITS) return;
    float b  = beta[i];
    float t  = atanhf(b);
    float qi = t * t + 0.1f;
    q[i] = qi;
    const float4* xr = (const float4*)(x + (size_t)i * D_DIM);
    float s = 0.f;
    #pragma unroll
    for (int c = 0; c < 4; ++c) {
        float4 v = xr[c];
        s += v.x * v.x + v.y * v.y + v.z * v.z + v.w * v.w;
    }
    xn[i] = s;
    int o = oid[i];
    if (o >= 1) {
        // pack: max over q (positive float -> monotone as uint bits);
        // tie-break to smallest index via inverted low word (matches jnp.argmax)
        unsigned long long pk = ((unsigned long long)__float_as_uint(qi) << 32)
                              | (unsigned long long)(0xFFFFFFFFu - (unsigned)i);
        atomicMax(&packed[o - 1], pk);
        atomicAdd(&cnt[o - 1], 1u);
    }
}

// ---------------- kernel 3: gather condensation points ----------------
__global__ void k_gather(const float* __restrict__ x,
                         const unsigned long long* __restrict__ packed,
                         const unsigned int* __restrict__ cnt,
                         const float* __restrict__ q, const float* __restrict__ xn,
                         unsigned int* __restrict__ alpha,
                         float* __restrict__ xk, float* __restrict__ qk,
                         float* __restrict__ xkn,
                         float* __restrict__ inv_att, float* __restrict__ inv_rep) {
    int k = blockIdx.x * blockDim.x + threadIdx.x;
    if (k >= K_PAD) return;
    if (k < K_OBJ) {
        unsigned long long pk = packed[k];
        unsigned idx = 0xFFFFFFFFu - (unsigned)(pk & 0xFFFFFFFFull);
        alpha[k] = idx;
        qk[k]  = q[idx];
        xkn[k] = xn[idx];
        #pragma unroll
        for (int d = 0; d < D_DIM; ++d) xk[k * D_DIM + d] = x[(size_t)idx * D_DIM + d];
        float c = (float)cnt[k];
        inv_att[k] = 1.f / ((c + EPSF) * (float)K_OBJ);
        inv_rep[k] = 1.f / (((float)N_HITS - c + EPSF) * (float)K_OBJ);
    } else {
        alpha[k] = 0u; qk[k] = 0.f; xkn[k] = 0.f;
        inv_att[k] = 0.f; inv_rep[k] = 0.f;
        #pragma unroll
        for (int d = 0; d < D_DIM; ++d) xk[k * D_DIM + d] = 0.f;
    }
}

// ---------------- kernel 4: N x K pair grid via fp32 WMMA ----------------
__global__ void __launch_bounds__(256)
k_pair(const float* __restrict__ x,
       const int*   __restrict__ oid,
       const float* __restrict__ wgt,
       const float* __restrict__ q,  const float* __restrict__ xn,
       const float* __restrict__ xk, const float* __restrict__ qk,
       const float* __restrict__ xkn,
       const float* __restrict__ inv_att, const float* __restrict__ inv_rep,
       float* __restrict__ att_part, float* __restrict__ rep_part) {
    // LDS staging of all column-side data (CDNA5: 320KB per WGP)
    __shared__ float s_xk[K_PAD * XK_STRIDE]; // padded stride -> conflict-free b64 reads
    __shared__ float s_ca[K_PAD];             // q_k * inv_att
    __shared__ float s_cr[K_PAD];             // q_k * inv_rep
    __shared__ float s_xkn[K_PAD];            // ||x_k||^2
    __shared__ float s_reda[256];
    __shared__ float s_redr[256];

    const int tid   = threadIdx.x;
    const int lane  = tid & 31;
    const int wave  = tid >> 5;
    const int col   = lane & 15;   // A-row / B-col / C-col for this lane
    const int hi    = lane >> 4;   // lane-group select
    const int gwave = blockIdx.x * (blockDim.x >> 5) + wave;
    const int nwav  = gridDim.x * (blockDim.x >> 5);

    for (int t = tid; t < K_PAD; t += blockDim.x) {
        float qq  = qk[t];
        s_ca[t]   = qq * inv_att[t];
        s_cr[t]   = qq * inv_rep[t];
        s_xkn[t]  = xkn[t];
        const float4* src = (const float4*)(xk + (size_t)t * D_DIM);
        #pragma unroll
        for (int c = 0; c < 4; ++c) {
            float4 v = src[c];
            s_xk[t * XK_STRIDE + 4 * c + 0] = v.x;
            s_xk[t * XK_STRIDE + 4 * c + 1] = v.y;
            s_xk[t * XK_STRIDE + 4 * c + 2] = v.z;
            s_xk[t * XK_STRIDE + 4 * c + 3] = v.w;
        }
    }
    __syncthreads();

    float att_acc = 0.f, rep_acc = 0.f;

    for (int rt = gwave; rt < ROW_TILES; rt += nwav) {
        const int i0 = rt * 16;
        // A fragments (16x4 f32): lane row = col, dims = 4c + 2*hi + {0,1}
        v2f a[4];
        const float* xr = x + (size_t)(i0 + col) * D_DIM + 2 * hi;
        #pragma unroll
        for (int c = 0; c < 4; ++c) { a[c].x = xr[4 * c]; a[c].y = xr[4 * c + 1]; }
        // row scalars for this thread's C rows: i = i0 + 8*hi + r
        float rwq[8], rxn[8]; int ro[8];
        #pragma unroll
        for (int r = 0; r < 8; ++r) {
            int i = i0 + 8 * hi + r;
            rwq[r] = wgt[i] * q[i];    // folded per-row coefficient
            rxn[r] = xn[i];
            ro[r]  = oid[i];
        }
        for (int ct = 0; ct < COL_TILES; ++ct) {
            const int n = ct * 16 + col;
            // B fragments (4x16 f32): B[d][n] = xk[n][d], d = 4c + 2*hi + {0,1}
            v2f b[4];
            const float* xc = s_xk + n * XK_STRIDE + 2 * hi;
            #pragma unroll
            for (int c = 0; c < 4; ++c) { b[c].x = xc[4 * c]; b[c].y = xc[4 * c + 1]; }
            const float ca = s_ca[n], cr = s_cr[n], cxn = s_xkn[n];

            v8f acc = {};
            #pragma unroll
            for (int c = 0; c < 4; ++c)
                acc = __builtin_amdgcn_wmma_f32_16x16x4_f32(
                        false, a[c], false, b[c], (short)0, acc, false, false);

            #pragma unroll
            for (int r = 0; r < 8; ++r) {
                float dot  = acc[r];
                float d2   = fmaxf(rxn[r] + cxn - 2.f * dot, 0.f);
                float dist = __builtin_amdgcn_sqrtf(d2 + 1e-12f);
                float wq   = rwq[r];
                float att_t = wq * ca * d2;
                float rep_t = wq * cr * (1.f - dist);
                bool  is_att = (ro[r] == n + 1);
                att_acc += is_att ? att_t : 0.f;                       // cndmask, no branch
                rep_acc += (!is_att && (dist < 1.f)) ? rep_t : 0.f;    // cndmask, no branch
            }
        }
    }

    s_reda[tid] = att_acc; s_redr[tid] = rep_acc;
    __syncthreads();
    for (int off = 128; off > 0; off >>= 1) {
        if (tid < off) { s_reda[tid] += s_reda[tid + off]; s_redr[tid] += s_redr[tid + off]; }
        __syncthreads();
    }
    if (tid == 0) { att_part[blockIdx.x] = s_reda[0]; rep_part[blockIdx.x] = s_redr[0]; }
}

// ---------------- kernel 5: deterministic finalize ----------------
__global__ void k_final(const float* __restrict__ beta,
                        const int*   __restrict__ oid,
                        const unsigned int* __restrict__ alpha,
                        const float* __restrict__ att_part,
                        const float* __restrict__ rep_part,
                        float* __restrict__ out) {
    const int tid = threadIdx.x;
    float a = 0.f, r = 0.f, ns = 0.f, nc = 0.f, cw = 0.f;
    for (int j = tid; j < NB_PAIR; j += 256) { a += att_part[j]; r += rep_part[j]; }
    for (int i = tid; i < N_HITS; i += 256)
        if (oid[i] <= 0) { ns += beta[i]; nc += 1.f; }
    for (int k = tid; k < K_OBJ; k += 256) cw += 1.f - beta[alpha[k]];

    __shared__ float sa[256], sr[256], sn[256], sc[256], sw[256];
    sa[tid] = a; sr[tid] = r; sn[tid] = ns; sc[tid] = nc; sw[tid] = cw;
    __syncthreads();
    for (int off = 128; off > 0; off >>= 1) {
        if (tid < off) {
            sa[tid] += sa[tid + off]; sr[tid] += sr[tid + off];
            sn[tid] += sn[tid + off]; sc[tid] += sc[tid + off];
            sw[tid] += sw[tid + off];
        }
        __syncthreads();
    }
    if (tid == 0) {
        out[0] = sa[0];                       // v_att
        out[1] = sr[0];                       // v_rep
        out[2] = sw[0] / (float)K_OBJ;        // l_coward
        out[3] = sn[0] / sc[0];               // l_noise
    }
}

extern "C" void kernel_launch(void* const* d_in, const int* in_sizes, int n_in,
                              void* d_out, int out_size, void* d_ws, size_t ws_size,
                              hipStream_t stream) {
    const float* beta = (const float*)d_in[0];
    const float* x    = (const float*)d_in[1];
    const float* wgt  = (const float*)d_in[2];
    const int*   oid  = (const int*)  d_in[3];
    float* out = (float*)d_out;
    WS w = carve(d_ws);

    k_init<<<(K_PAD + 255) / 256, 256, 0, stream>>>(w.packed, w.cnt);
    k_hit<<<(N_HITS + 255) / 256, 256, 0, stream>>>(beta, x, oid, w.packed, w.cnt, w.q, w.xn);
    k_gather<<<(K_PAD + 255) / 256, 256, 0, stream>>>(x, w.packed, w.cnt, w.q, w.xn,
                                                      w.alpha, w.xk, w.qk, w.xkn,
                                                      w.inv_att, w.inv_rep);
    k_pair<<<NB_PAIR, 256, 0, stream>>>(x, oid, wgt, w.q, w.xn, w.xk, w.qk, w.xkn,
                                        w.inv_att, w.inv_rep, w.att_part, w.rep_part);
    k_final<<<1, 256, 0, stream>>>(beta, oid, w.alpha, w.att_part, w.rep_part, out);
}